// RBLNGptOssTopKRouter_46231027974602
// MI455X (gfx1250) — compile-verified
//
#include <hip/hip_runtime.h>
#include <hip/hip_bf16.h>
#include <math.h>

// GPT-OSS Top-K router: logits = x @ W^T + bias ; top-4 ; softmax(top4) ; scatter.
// x: [T=8192, D=2880] f32, W: [E=32, D] f32, bias: [E] f32.
// d_out = [T*E] f32 scores followed by [T*4] int32 indices.
//
// Memory-bound (94 MB x-stream, AI ~ 16 F/B => ~4 us at 23.3 TB/s).
// fp32 WMMA v_wmma_f32_16x16x4_f32; 4 waves split the K=2880 reduction per
// 16-token tile (2048 waves total) and each wave keeps ~45 b64 loads in
// flight (unroll 15) => ~22 MB aggregate in-flight, matching the HBM
// bandwidth-delay product. Partials are reduced deterministically via LDS.

typedef __attribute__((ext_vector_type(2))) float v2f;
typedef __attribute__((ext_vector_type(8))) float v8f;

#define ROUTER_D 2880
#define ROUTER_E 32
#define ROUTER_K 4
#define KCHUNKS 4                         // waves cooperating on one tile
#define TILES_PER_BLOCK 2
#define WAVES_PER_BLOCK (KCHUNKS * TILES_PER_BLOCK)   // 8 waves = 256 threads
#define CHUNK_ELEMS (ROUTER_D / KCHUNKS)  // 720 K-elements per wave

__global__ __launch_bounds__(256)
void gptoss_topk_router_kernel(const float* __restrict__ x,
                               const float* __restrict__ weight,
                               const float* __restrict__ bias,
                               float* __restrict__ scores,
                               int* __restrict__ idx_out,
                               int T)
{
    // partial 16x32 logit tiles, one per (tile, k-chunk); stride 33 avoids
    // LDS bank conflicts on the per-token reduction reads
    __shared__ float partials[TILES_PER_BLOCK][KCHUNKS][16][ROUTER_E + 1];

    const int lane = threadIdx.x & 31;
    const int wave = threadIdx.x >> 5;
    const int tsub = wave >> 2;                 // tile within block
    const int kc   = wave & (KCHUNKS - 1);      // K-chunk owned by this wave
    const int tile = blockIdx.x * TILES_PER_BLOCK + tsub;
    const int tok0 = tile * 16;

    const int l15 = lane & 15;   // A: row M ; B: expert column N
    const int h   = lane >> 4;   // half-wave selects K sub-pair {0,1} vs {2,3}

    int arow = tok0 + l15;
    if (arow >= T) arow = T - 1; // clamp ragged tail; writes are guarded below

    const int k0 = kc * CHUNK_ELEMS;

    // ISA 32-bit A layout (16x4): lane L, VGPR v -> M = L%16, K = 2*(L>>4)+v
    const float* aptr  = x      + (size_t)arow       * ROUTER_D + k0 + 2 * h;
    // B layout (4x16): lane L, VGPR v -> N = L%16, K = 2*(L>>4)+v ; B[k][n]=W[n][k]
    const float* b0ptr = weight + (size_t)l15        * ROUTER_D + k0 + 2 * h;
    const float* b1ptr = weight + (size_t)(l15 + 16) * ROUTER_D + k0 + 2 * h;

    v8f acc0 = {}; // experts 0..15
    v8f acc1 = {}; // experts 16..31

    // 180 k-steps; unroll 15 -> ~45 b64 loads in flight per wave
    #pragma unroll 15
    for (int k = 0; k < CHUNK_ELEMS; k += 4) {
        v2f a  = *(const v2f*)(aptr  + k);
        v2f b0 = *(const v2f*)(b0ptr + k);
        v2f b1 = *(const v2f*)(b1ptr + k);
        acc0 = __builtin_amdgcn_wmma_f32_16x16x4_f32(
                   false, a, false, b0, (short)0, acc0, false, false);
        acc1 = __builtin_amdgcn_wmma_f32_16x16x4_f32(
                   false, a, false, b1, (short)0, acc1, false, false);
    }

    // C/D layout: lane L, VGPR v -> M = v + 8*(L>>4), N = L%16
    #pragma unroll
    for (int v = 0; v < 8; ++v) {
        const int row = v + 8 * h;
        partials[tsub][kc][row][l15]      = acc0[v];
        partials[tsub][kc][row][l15 + 16] = acc1[v];
    }

    __syncthreads();

    // One wave per tile (kc==0), lanes 0..15: reduce 4 partials + bias,
    // top-4 (first-index tie-break like jax.lax.top_k), softmax, scatter.
    if (kc == 0 && lane < 16) {
        const int tok = tok0 + lane;
        if (tok < T) {
            float lg[ROUTER_E];
            #pragma unroll
            for (int j = 0; j < ROUTER_E; ++j) {
                float s = partials[tsub][0][lane][j];
                s += partials[tsub][1][lane][j];
                s += partials[tsub][2][lane][j];
                s += partials[tsub][3][lane][j];
                lg[j] = s + bias[j];
            }

            int   sel[ROUTER_K];
            float sv[ROUTER_K];
            unsigned taken = 0u;
            #pragma unroll
            for (int kk = 0; kk < ROUTER_K; ++kk) {
                float best = -INFINITY;
                int   bi   = 0;
                #pragma unroll
                for (int j = 0; j < ROUTER_E; ++j) {
                    if (!((taken >> j) & 1u) && lg[j] > best) { best = lg[j]; bi = j; }
                }
                sel[kk] = bi; sv[kk] = best; taken |= (1u << bi);
            }

            // softmax over the 4 selected (sv[0] is the max)
            float m = sv[0], sum = 0.f;
            #pragma unroll
            for (int kk = 0; kk < ROUTER_K; ++kk) { sv[kk] = __expf(sv[kk] - m); sum += sv[kk]; }
            const float inv = 1.f / sum;

            float* srow = scores + (size_t)tok * ROUTER_E; // 128B-aligned row
            const float4 z4 = make_float4(0.f, 0.f, 0.f, 0.f);
            #pragma unroll
            for (int j = 0; j < ROUTER_E; j += 4)
                *(float4*)(srow + j) = z4;
            #pragma unroll
            for (int kk = 0; kk < ROUTER_K; ++kk)  // same-wave stores stay ordered
                srow[sel[kk]] = sv[kk] * inv;

            int* irow = idx_out + (size_t)tok * ROUTER_K;
            #pragma unroll
            for (int kk = 0; kk < ROUTER_K; ++kk)
                irow[kk] = sel[kk];
        }
    }
}

extern "C" void kernel_launch(void* const* d_in, const int* in_sizes, int n_in,
                              void* d_out, int out_size, void* d_ws, size_t ws_size,
                              hipStream_t stream) {
    (void)n_in; (void)out_size; (void)d_ws; (void)ws_size;
    const float* x      = (const float*)d_in[0];
    const float* weight = (const float*)d_in[1];
    const float* bias   = (const float*)d_in[2];

    const int T = in_sizes[0] / ROUTER_D;      // 8192 tokens
    float* scores = (float*)d_out;             // [T, 32] f32
    int*   idxs   = (int*)d_out + (size_t)T * ROUTER_E; // [T, 4] int32 after scores

    const int tiles  = (T + 15) / 16;                                    // 512
    const int blocks = (tiles + TILES_PER_BLOCK - 1) / TILES_PER_BLOCK;  // 256
    gptoss_topk_router_kernel<<<blocks, 32 * WAVES_PER_BLOCK, 0, stream>>>(
        x, weight, bias, scores, idxs, T);
}